// MoELoRALayer_25623774888286
// MI455X (gfx1250) — compile-verified
//
#include <hip/hip_runtime.h>

typedef __attribute__((ext_vector_type(2))) float v2f;
typedef __attribute__((ext_vector_type(8))) float v8f;

#define B_ 128
#define T_ 197
#define D_ 768
#define E_ 8
#define R_ 8
#define O_ 2304
#define MT 13    // ceil(197/16)
#define NT 144   // 2304/16

// ---------------------------------------------------------------------------
// Kernel 1: router — pooled mean, logits, softmax, top1/argmax, per-batch outs
// ---------------------------------------------------------------------------
__global__ __launch_bounds__(256) void router_kernel(
    const float* __restrict__ tokens, const float* __restrict__ w_gate,
    const float* __restrict__ scaling,
    float* __restrict__ out_logits, float* __restrict__ out_sel,
    float* __restrict__ out_ew,
    float* __restrict__ ws_gscale, float* __restrict__ ws_probs,
    int* __restrict__ ws_sel)
{
    __shared__ float pooled[D_];
    __shared__ float red[256][E_];
    const int b   = blockIdx.x;
    const int tid = threadIdx.x;
    const float* tb = tokens + (size_t)b * T_ * D_;

    // mean over T (coalesced across d)
    for (int d = tid; d < D_; d += 256) {
        float s = 0.f;
        for (int t = 0; t < T_; ++t) s += tb[(size_t)t * D_ + d];
        pooled[d] = s * (1.0f / (float)T_);
    }
    __syncthreads();

    // logits = pooled @ w_gate
    float le[E_];
    #pragma unroll
    for (int e = 0; e < E_; ++e) le[e] = 0.f;
    for (int d = tid; d < D_; d += 256) {
        float p = pooled[d];
        #pragma unroll
        for (int e = 0; e < E_; ++e) le[e] += p * w_gate[d * E_ + e];
    }
    #pragma unroll
    for (int e = 0; e < E_; ++e) red[tid][e] = le[e];
    __syncthreads();
    for (int off = 128; off > 0; off >>= 1) {
        if (tid < off) {
            #pragma unroll
            for (int e = 0; e < E_; ++e) red[tid][e] += red[tid + off][e];
        }
        __syncthreads();
    }

    if (tid == 0) {
        float logit[E_], prob[E_];
        float mx = -1e30f;
        #pragma unroll
        for (int e = 0; e < E_; ++e) { logit[e] = red[0][e]; mx = fmaxf(mx, logit[e]); }
        float se = 0.f;
        #pragma unroll
        for (int e = 0; e < E_; ++e) { prob[e] = expf(logit[e] - mx); se += prob[e]; }
        float inv = 1.f / se;
        int   sel = 0;
        float top1 = -1.f;
        #pragma unroll
        for (int e = 0; e < E_; ++e) {
            prob[e] *= inv;
            if (prob[e] > top1) { top1 = prob[e]; sel = e; }   // first max on ties
        }
        #pragma unroll
        for (int e = 0; e < E_; ++e) {
            out_logits[b * E_ + e] = logit[e];
            ws_probs[b * E_ + e]   = prob[e];
            out_ew[b * E_ + e]     = (e == sel) ? top1 : 0.f;
        }
        out_sel[b]   = (float)sel;
        ws_sel[b]    = sel;
        ws_gscale[b] = scaling[sel] * top1;
    }
}

// ---------------------------------------------------------------------------
// Kernel 2: deterministic reduction for importance / load (no atomics)
// ---------------------------------------------------------------------------
__global__ void stats_kernel(const float* __restrict__ ws_probs,
                             const float* __restrict__ out_ew,
                             float* __restrict__ out_imp,
                             float* __restrict__ out_load)
{
    int e = threadIdx.x;
    if (e < E_) {
        float imp = 0.f, ld = 0.f;
        for (int b = 0; b < B_; ++b) {
            imp += out_ew[b * E_ + e];
            ld  += ws_probs[b * E_ + e];
        }
        out_imp[e]  = imp;
        out_load[e] = ld;
    }
}

// ---------------------------------------------------------------------------
// Kernel 3: h = tokens @ lora_a[sel]  via V_WMMA_F32_16X16X4_F32
// one wave per (16-row token tile, batch); N padded 8->16 with zero mask
// ---------------------------------------------------------------------------
__global__ __launch_bounds__(32) void lora_a_kernel(
    const float* __restrict__ tokens, const float* __restrict__ lora_a,
    const int* __restrict__ ws_sel, float* __restrict__ ws_h)
{
    const int tile = blockIdx.x;      // M tile 0..12
    const int b    = blockIdx.y;      // batch
    const int l    = threadIdx.x;     // lane 0..31
    const int half = l >> 4;          // 0 | 1
    const int lm   = l & 15;
    const int e    = ws_sel[b];
    const float* Ae = lora_a + (size_t)e * D_ * R_;

    // A-frag: lane holds token row M=lm, K = 2*half + {0,1}
    const int   row   = tile * 16 + lm;
    const float rmask = (row < T_) ? 1.f : 0.f;
    const int   rowc  = (row < T_) ? row : (T_ - 1);         // clamp: in-bounds addr
    const float* trow = tokens + ((size_t)b * T_ + rowc) * D_;

    // B-frag: lane holds col N=lm (valid only N<8), K = 2*half + {0,1}
    const float nmask = (lm < R_) ? 1.f : 0.f;
    const int   ncl   = (lm < R_) ? lm : (R_ - 1);
    const int   koff  = half * 2;

    v8f c = {0.f, 0.f, 0.f, 0.f, 0.f, 0.f, 0.f, 0.f};
    for (int k = 0; k < D_; k += 4) {
        const int k0 = k + koff;
        v2f a, bm;
        a.x  = trow[k0]     * rmask;
        a.y  = trow[k0 + 1] * rmask;
        bm.x = Ae[(size_t)k0 * R_ + ncl]       * nmask;
        bm.y = Ae[(size_t)(k0 + 1) * R_ + ncl] * nmask;
        c = __builtin_amdgcn_wmma_f32_16x16x4_f32(false, a, false, bm,
                                                  (short)0, c, false, false);
    }

    // C layout: lane lm = col N (r index), rows M = v + 8*half
    if (lm < R_) {
        #pragma unroll
        for (int v = 0; v < 8; ++v) {
            int gm = tile * 16 + v + 8 * half;
            if (gm < T_) ws_h[((size_t)b * T_ + gm) * R_ + lm] = c[v];
        }
    }
}

// ---------------------------------------------------------------------------
// Kernel 4: upd = (h @ lora_b[sel]) * gate_scale  via V_WMMA_F32_16X16X4_F32
// one wave per (8 N-tiles, M-tile, batch); A-frags (h) reused across N-tiles
// ---------------------------------------------------------------------------
__global__ __launch_bounds__(32) void lora_b_kernel(
    const float* __restrict__ ws_h, const float* __restrict__ lora_b,
    const int* __restrict__ ws_sel, const float* __restrict__ ws_gscale,
    float* __restrict__ out_upd)
{
    const int nt0  = blockIdx.x * 8;  // first N tile handled by this wave
    const int tile = blockIdx.y;      // M tile
    const int b    = blockIdx.z;      // batch
    const int l    = threadIdx.x;
    const int half = l >> 4;
    const int lm   = l & 15;
    const int e    = ws_sel[b];
    const float gs = ws_gscale[b];
    const float* Be = lora_b + (size_t)e * R_ * O_;

    // A-frags from h: K=8 split into two 16x4 frags
    const int   row   = tile * 16 + lm;
    const float rmask = (row < T_) ? 1.f : 0.f;
    const int   rowc  = (row < T_) ? row : (T_ - 1);
    const float* hrow = ws_h + ((size_t)b * T_ + rowc) * R_;
    const int   koff  = half * 2;

    v2f a0, a1;
    a0.x = hrow[koff]         * rmask;
    a0.y = hrow[koff + 1]     * rmask;
    a1.x = hrow[4 + koff]     * rmask;
    a1.y = hrow[5 + koff]     * rmask;

    for (int i = 0; i < 8; ++i) {
        const int col = (nt0 + i) * 16 + lm;   // O_=2304 is a multiple of 16
        v2f b0, b1;
        b0.x = Be[(size_t)(koff)     * O_ + col];
        b0.y = Be[(size_t)(koff + 1) * O_ + col];
        b1.x = Be[(size_t)(koff + 4) * O_ + col];
        b1.y = Be[(size_t)(koff + 5) * O_ + col];

        v8f c = {0.f, 0.f, 0.f, 0.f, 0.f, 0.f, 0.f, 0.f};
        c = __builtin_amdgcn_wmma_f32_16x16x4_f32(false, a0, false, b0,
                                                  (short)0, c, false, false);
        c = __builtin_amdgcn_wmma_f32_16x16x4_f32(false, a1, false, b1,
                                                  (short)0, c, false, false);

        // store 16x16 tile: lanes 0-15 cover 16 consecutive cols (64B) per VGPR
        #pragma unroll
        for (int v = 0; v < 8; ++v) {
            int gm = tile * 16 + v + 8 * half;
            if (gm < T_)
                out_upd[((size_t)b * T_ + gm) * O_ + col] = c[v] * gs;
        }
    }
}

// ---------------------------------------------------------------------------
extern "C" void kernel_launch(void* const* d_in, const int* in_sizes, int n_in,
                              void* d_out, int out_size, void* d_ws, size_t ws_size,
                              hipStream_t stream)
{
    const float* tokens  = (const float*)d_in[0];
    const float* w_gate  = (const float*)d_in[1];
    const float* lora_a  = (const float*)d_in[2];
    const float* lora_b  = (const float*)d_in[3];
    const float* scaling = (const float*)d_in[4];

    // d_out layout (reference tuple order, flattened):
    float* out        = (float*)d_out;
    float* out_upd    = out;                                  // B*T*O
    float* out_logits = out + (size_t)B_ * T_ * O_;           // B*E
    float* out_sel    = out_logits + B_ * E_;                 // B
    float* out_ew     = out_sel + B_;                         // B*E
    float* out_imp    = out_ew + B_ * E_;                     // E
    float* out_load   = out_imp + E_;                         // E

    // workspace layout
    float* ws        = (float*)d_ws;
    float* ws_gscale = ws;                         // 128 floats
    float* ws_probs  = ws + 128;                   // 1024 floats
    int*   ws_sel    = (int*)(ws + 128 + 1024);    // 128 ints
    float* ws_h      = ws + 128 + 1024 + 128;      // B*T*R floats (~807 KB)

    router_kernel<<<dim3(B_), dim3(256), 0, stream>>>(
        tokens, w_gate, scaling, out_logits, out_sel, out_ew,
        ws_gscale, ws_probs, ws_sel);

    stats_kernel<<<dim3(1), dim3(8), 0, stream>>>(
        ws_probs, out_ew, out_imp, out_load);

    lora_a_kernel<<<dim3(MT, B_), dim3(32), 0, stream>>>(
        tokens, lora_a, ws_sel, ws_h);

    lora_b_kernel<<<dim3(NT / 8, MT, B_), dim3(32), 0, stream>>>(
        ws_h, lora_b, ws_sel, ws_gscale, out_upd);
}